// SelectiveSSM_6828998000923
// MI455X (gfx1250) — compile-verified
//
#include <hip/hip_runtime.h>
#include <hip/hip_bf16.h>
#include <math.h>
#include <stdint.h>

// Problem constants (fixed by the reference's setup_inputs()).
#define DT_RANK 64
#define D_STATE 16
#define BATCH   2
#define LSEQ    2048
#define DIM     2048
#define NPROJ   96                 // DT_RANK + 2*D_STATE
#define M_TOTAL (BATCH * LSEQ)     // 4096 rows in the flattened (b,l) dim

typedef float v2f __attribute__((ext_vector_type(2)));
typedef float v8f __attribute__((ext_vector_type(8)));

__device__ __forceinline__ v2f ld2(const float* p) { return *(const v2f*)p; }

__device__ __forceinline__ v8f wmma_f32_4(v2f a, v2f b, v8f c) {
    // V_WMMA_F32_16X16X4_F32 : D = A(16x4,f32) * B(4x16,f32) + C(16x16,f32)
    return __builtin_amdgcn_wmma_f32_16x16x4_f32(false, a, false, b,
                                                 (short)0, c, false, false);
}

// Async copy of one 16-byte segment global -> LDS (tracked by ASYNCcnt).
__device__ __forceinline__ void async_g2lds_b128(uint32_t lds_off, const void* g) {
    uint64_t ga = (uint64_t)(uintptr_t)g;
    asm volatile("global_load_async_to_lds_b128 %0, %1, off"
                 :: "v"(lds_off), "v"(ga) : "memory");
}
__device__ __forceinline__ void wait_asynccnt0() {
    asm volatile("s_wait_asynccnt 0x0" ::: "memory");
}

// ---------------------------------------------------------------------------
// Kernel 1: x_dbl[bl][e] = sum_d x[bl][d] * W_xproj[e][d]
// M=4096, N=96 (6 tiles), K=2048. One wave per 16-row M-tile; the A fragment
// is loaded once per K-step and reused across all 6 N-tiles.
// ---------------------------------------------------------------------------
__global__ void __launch_bounds__(32)
k_proj(const float* __restrict__ x, const float* __restrict__ Wxp,
       float* __restrict__ xdbl) {
    const int mt   = blockIdx.x;        // 0..255
    const int lane = threadIdx.x;       // 0..31
    const int lnlo = lane & 15;
    const int lnhi = lane >> 4;

    const float* arow = x + (size_t)(mt * 16 + lnlo) * DIM + 2 * lnhi;

    v8f acc[6] = {};
    for (int k0 = 0; k0 < DIM; k0 += 4) {
        v2f a = ld2(arow + k0);
        #pragma unroll
        for (int j = 0; j < 6; ++j) {
            v2f b = ld2(Wxp + (size_t)(j * 16 + lnlo) * DIM + k0 + 2 * lnhi);
            acc[j] = wmma_f32_4(a, b, acc[j]);
        }
    }
    #pragma unroll
    for (int j = 0; j < 6; ++j) {
        #pragma unroll
        for (int r = 0; r < 8; ++r) {
            const int m = r + 8 * lnhi;     // C/D layout: VGPR r, lane half
            const int n = lnlo;
            xdbl[(size_t)(mt * 16 + m) * NPROJ + j * 16 + n] = acc[j][r];
        }
    }
}

// ---------------------------------------------------------------------------
// Kernel 2: delta[bl][d] = softplus( sum_r x_dbl[bl][r] * W_dt[d][r] + b_dt[d] )
// M=4096, N=2048, K=64. One wave per (M-tile, group of 8 N-tiles).
// The full 16x64 A tile (16 float2 frags) is preloaded into VGPRs.
// ---------------------------------------------------------------------------
__global__ void __launch_bounds__(32)
k_delta(const float* __restrict__ xdbl, const float* __restrict__ Wdt,
        const float* __restrict__ bdt, float* __restrict__ delta) {
    const int bid  = blockIdx.x;        // 0..4095
    const int mt   = bid >> 4;          // 0..255
    const int ng   = bid & 15;          // 0..15  -> N-tiles ng*8 .. ng*8+7
    const int lane = threadIdx.x;
    const int lnlo = lane & 15;
    const int lnhi = lane >> 4;

    const float* arow = xdbl + (size_t)(mt * 16 + lnlo) * NPROJ + 2 * lnhi;
    v2f afrag[16];
    #pragma unroll
    for (int s = 0; s < 16; ++s) afrag[s] = ld2(arow + 4 * s);

    v8f acc[8] = {};
    #pragma unroll
    for (int s = 0; s < 16; ++s) {
        #pragma unroll
        for (int j = 0; j < 8; ++j) {
            const int n0 = (ng * 8 + j) * 16;
            v2f b = ld2(Wdt + (size_t)(n0 + lnlo) * DT_RANK + 4 * s + 2 * lnhi);
            acc[j] = wmma_f32_4(afrag[s], b, acc[j]);
        }
    }
    #pragma unroll
    for (int j = 0; j < 8; ++j) {
        #pragma unroll
        for (int r = 0; r < 8; ++r) {
            const int m  = r + 8 * lnhi;
            const int bl = mt * 16 + m;
            const int d  = (ng * 8 + j) * 16 + lnlo;
            float z  = acc[j][r] + bdt[d];
            // numerically stable softplus
            float sp = fmaxf(z, 0.0f) + log1pf(__expf(-fabsf(z)));
            delta[(size_t)bl * DIM + d] = sp;
        }
    }
}

// ---------------------------------------------------------------------------
// Kernel 3: selective scan. One thread per (b,d) channel, h[16] in VGPRs.
// The per-step shared B/C vectors (32 floats/step) are streamed into LDS with
// GLOBAL_LOAD_ASYNC_TO_LDS_B128 (ASYNCcnt-tracked), double-buffered so the
// copy of chunk i+1 overlaps the 128 serial scan steps of chunk i.
// delta/x/out accesses are fully coalesced across the 64 d-channels.
// ---------------------------------------------------------------------------
#define SCAN_BLK 64
#define CHUNK    128

__device__ __forceinline__ void issue_bc_chunk(float* sbuf,
                                               const float* __restrict__ xdbl,
                                               size_t bl0, int l0, int tid) {
    // CHUNK rows of 32 floats (128 B) at row stride NPROJ floats.
    // 8 x b128 segments per row, CHUNK*8 = 1024 segments, 16 per thread.
    for (int seg = tid; seg < CHUNK * 8; seg += SCAN_BLK) {
        const int st = seg >> 3;
        const int c  = (seg & 7) * 4;                  // float index in row
        const uint32_t lds_off = (uint32_t)(uintptr_t)&sbuf[st * 32 + c];
        async_g2lds_b128(lds_off, &xdbl[(bl0 + l0 + st) * NPROJ + DT_RANK + c]);
    }
}

__global__ void __launch_bounds__(SCAN_BLK)
k_scan(const float* __restrict__ x, const float* __restrict__ xdbl,
       const float* __restrict__ delta, const float* __restrict__ Alog,
       const float* __restrict__ Dp, float* __restrict__ out) {
    __shared__ float sBC[2][CHUNK * 32];   // per step: [0..15]=B, [16..31]=C

    const int b      = blockIdx.x >> 5;          // DIM/SCAN_BLK = 32 blocks/batch
    const int dchunk = blockIdx.x & 31;
    const int tid    = threadIdx.x;
    const int d      = dchunk * SCAN_BLK + tid;

    float A[D_STATE], h[D_STATE];
    #pragma unroll
    for (int n = 0; n < D_STATE; ++n) {
        A[n] = -__expf(Alog[(size_t)d * D_STATE + n]);
        h[n] = 0.0f;
    }
    const float dpar = Dp[d];
    const size_t bl0 = (size_t)b * LSEQ;

    issue_bc_chunk(sBC[0], xdbl, bl0, 0, tid);       // prime buffer 0

    int buf = 0;
    for (int l0 = 0; l0 < LSEQ; l0 += CHUNK, buf ^= 1) {
        wait_asynccnt0();        // this wave's async copies into sBC[buf] done
        __syncthreads();         // all waves done copying + done reading buf^1

        if (l0 + CHUNK < LSEQ) {
            issue_bc_chunk(sBC[buf ^ 1], xdbl, bl0, l0 + CHUNK, tid);
            // hint next chunk of delta/x toward caches (global_prefetch_b8)
            __builtin_prefetch(&delta[(bl0 + l0 + CHUNK) * DIM + d], 0, 1);
            __builtin_prefetch(&x[(bl0 + l0 + CHUNK) * DIM + d], 0, 1);
        }

        const float* sbuf = sBC[buf];
        for (int l = l0; l < l0 + CHUNK; ++l) {
            const size_t bl = bl0 + l;
            const float dl  = delta[bl * DIM + d];
            const float xv  = x[bl * DIM + d];
            const float dbu = dl * xv;
            const float* bc = &sbuf[(l - l0) * 32];
            float y = 0.0f;
            #pragma unroll
            for (int n = 0; n < D_STATE; ++n) {
                float dA = __expf(dl * A[n]);      // v_exp_f32, 16 indep chains
                h[n] = dA * h[n] + dbu * bc[n];
                y   += h[n] * bc[16 + n];
            }
            out[bl * DIM + d] = y + xv * dpar;
        }
    }
}

// ---------------------------------------------------------------------------
extern "C" void kernel_launch(void* const* d_in, const int* in_sizes, int n_in,
                              void* d_out, int out_size, void* d_ws, size_t ws_size,
                              hipStream_t stream) {
    (void)in_sizes; (void)n_in; (void)out_size; (void)ws_size;

    const float* x    = (const float*)d_in[0];   // (2,2048,2048)
    const float* Wxp  = (const float*)d_in[1];   // (96,2048)
    const float* Wdt  = (const float*)d_in[2];   // (2048,64)
    const float* bdt  = (const float*)d_in[3];   // (2048,)
    const float* Alog = (const float*)d_in[4];   // (2048,16)
    const float* Dp   = (const float*)d_in[5];   // (2048,)
    float*       out  = (float*)d_out;           // (2,2048,2048)

    char*  ws    = (char*)d_ws;
    float* delta = (float*)ws;                                   // 4096*2048*4 = 32 MiB
    float* xdbl  = (float*)(ws + (size_t)M_TOTAL * DIM * 4);     // 4096*96*4  = 1.5 MiB

    k_proj <<<M_TOTAL / 16,                 32, 0, stream>>>(x, Wxp, xdbl);
    k_delta<<<(M_TOTAL / 16) * (DIM / 128), 32, 0, stream>>>(xdbl, Wdt, bdt, delta);
    k_scan <<<BATCH * (DIM / SCAN_BLK), SCAN_BLK, 0, stream>>>(x, xdbl, delta, Alog, Dp, out);
}